// Mamba2Bidirectional_49615462204123
// MI455X (gfx1250) — compile-verified
//
#include <hip/hip_runtime.h>
#include <cstddef>
#include <cstdint>

#define DMODEL   1024
#define DSTATE   128
#define DINNER   2048
#define HEADDIM  64
#define NHEADS   32
#define CONVDIM  2304
#define DINPROJ  4384
#define BATCH    2
#define SEQLEN   2048

typedef __attribute__((ext_vector_type(2))) float v2f;
typedef __attribute__((ext_vector_type(8))) float v8f;

#if defined(__has_builtin)
#if __has_builtin(__builtin_amdgcn_tensor_load_to_lds)
#define HAS_TDM6 1
#endif
#endif

#ifdef HAS_TDM6
typedef unsigned int u32x4 __attribute__((ext_vector_type(4)));
typedef int i32x8 __attribute__((ext_vector_type(8)));
typedef int i32x4 __attribute__((ext_vector_type(4)));

// Issue one TDM 2D tile load: `rows` rows of 32 floats, row stride
// k_stride_elems, into LDS at lds_off with a 4-DWORD pad after each
// 32-DWORD row (-> stride 36 floats, conflict-free for our frag reads).
__device__ __forceinline__ void tdm_load_tile(const float* gsrc, unsigned lds_off,
                                              int rows, int k_stride_elems) {
  const unsigned long long ga = (unsigned long long)(uintptr_t)gsrc;
  u32x4 g0;
  g0[0] = 1u;                                              // count=1 (valid user D#)
  g0[1] = lds_off;                                         // lds_addr (bytes)
  g0[2] = (unsigned)(ga & 0xffffffffu);                    // global_addr[31:0]
  g0[3] = (unsigned)((ga >> 32) & 0x01ffffffu) | (2u << 30); // addr[56:32] | type=2
  i32x8 g1;
  g1[0] = (2 << 16) | (1 << 20) | (4 << 22) | (3 << 25);   // 4B elems, pad_en, 32dw interval, 4dw pad
  g1[1] = (32 << 16);                                      // tensor_dim0 = 32 (lo16)
  g1[2] = (rows << 16);                                    // tensor_dim1 lo16 = rows
  g1[3] = (32 << 16);                                      // tile_dim0 = 32
  g1[4] = rows;                                            // tile_dim1 = rows, tile_dim2 = 0
  g1[5] = k_stride_elems;                                  // tensor_dim0_stride lo32
  g1[6] = 0;
  g1[7] = 0;
  const i32x4 z4 = {0, 0, 0, 0};
  const i32x8 z8 = {0, 0, 0, 0, 0, 0, 0, 0};
  __builtin_amdgcn_tensor_load_to_lds(g0, g1, z4, z4, z8, 0);
}
#endif

// ---------------------------------------------------------------------------
// GEMM: C[M,N] = A[M,K] @ W[N,K]^T  using fp32 WMMA 16x16x4.
// Block = 64x32 tile, 256 threads = 8 waves, each wave one 16x16 accumulator.
// Lrev > 0: logical row r=b*L+l reads physical row b*L+(L-1-l) (seq reversal).
// Tiles staged into stride-36 LDS rows either by TDM (padded DMA) or manually.
// ---------------------------------------------------------------------------
#define BM 64
#define BN 32
#define KT 32

__global__ __launch_bounds__(256)
void gemm_f32_wmma(const float* __restrict__ A, const float* __restrict__ W,
                   float* __restrict__ C, const int M, const int N, const int K,
                   const int Lrev) {
  __shared__ float As[BM][KT + 4];   // row stride 36 floats
  __shared__ float Bs[BN][KT + 4];   // row stride 36 floats (B[n][k])

  const int tid  = threadIdx.x;
  const int lane = tid & 31;
  const int wave = tid >> 5;
  const int wr   = wave & 3;         // 4 row tiles of 16
  const int wc   = wave >> 2;        // 2 col tiles of 16
  const int r0   = blockIdx.x * BM;
  const int n0   = blockIdx.y * BN;

  const int a_r = tid >> 3;          // 0..31
  const int a_k = (tid & 7) << 2;    // 0,4,...,28

  // Hoisted (possibly reversed) source row pointers for the manual path.
  int row0 = r0 + a_r;
  int row1 = row0 + 32;
  if (Lrev > 0) {
    const int b0 = row0 / Lrev;
    row0 = b0 * Lrev + (Lrev - 1 - (row0 - b0 * Lrev));
    const int b1 = row1 / Lrev;
    row1 = b1 * Lrev + (Lrev - 1 - (row1 - b1 * Lrev));
  }
  const float* ap0 = A + (size_t)row0 * K + a_k;
  const float* ap1 = A + (size_t)row1 * K + a_k;
  const float* wp0 = W + (size_t)(n0 + (tid >> 3)) * K + a_k;

#ifdef HAS_TDM6
  const bool use_tdm = (Lrev == 0);
  const unsigned lds_off_A = (unsigned)(uintptr_t)&As[0][0];
  const unsigned lds_off_B = (unsigned)(uintptr_t)&Bs[0][0];
#endif

  v8f acc = {};

  for (int k0 = 0; k0 < K; k0 += KT) {
#ifdef HAS_TDM6
    if (use_tdm) {
      if (tid == 0) {
        tdm_load_tile(A + (size_t)r0 * K + k0, lds_off_A, BM, K);
        tdm_load_tile(W + (size_t)n0 * K + k0, lds_off_B, BN, K);
        __builtin_amdgcn_s_wait_tensorcnt(0);
      }
    } else
#endif
    {
      // ---- manual staging: A tile (64 x 32), B tile (32 x 32) ----
      *(float4*)&As[a_r][a_k]      = *(const float4*)(ap0 + k0);
      *(float4*)&As[a_r + 32][a_k] = *(const float4*)(ap1 + k0);
      *(float4*)&Bs[tid >> 3][a_k] = *(const float4*)(wp0 + k0);
      if (k0 + KT < K) __builtin_prefetch(wp0 + k0 + KT, 0, 1);
    }
    __syncthreads();

    // ---- 8 WMMA steps of K=4 each ----
    const int arow = wr * 16 + (lane & 15);
    const int koff = (lane >> 4) << 1;          // lanes 16-31 take K+2,K+3
    const int bn   = wc * 16 + (lane & 15);
#pragma unroll
    for (int kk = 0; kk < KT; kk += 4) {
      const v2f af = *(const v2f*)&As[arow][kk + koff];
      const v2f bf = *(const v2f*)&Bs[bn][kk + koff];
      acc = __builtin_amdgcn_wmma_f32_16x16x4_f32(
          /*neg_a=*/false, af, /*neg_b=*/false, bf,
          /*c_mod=*/(short)0, acc, /*reuse_a=*/false, /*reuse_b=*/false);
    }
    __syncthreads();
  }

  // C/D layout: VGPR i -> lanes 0-15: (M=i, N=lane), lanes 16-31: (M=8+i, N=lane-16)
  const int crow = r0 + wr * 16 + ((lane >> 4) << 3);
  const int ccol = n0 + wc * 16 + (lane & 15);
#pragma unroll
  for (int i = 0; i < 8; ++i)
    C[(size_t)(crow + i) * N + ccol] = acc[i];
}

// ---------------------------------------------------------------------------
// Depthwise causal conv (width 4) + bias + SiLU over the xBC slice of zxbcdt.
// ---------------------------------------------------------------------------
__global__ __launch_bounds__(256)
void conv_silu(const float* __restrict__ zx, const float* __restrict__ cw,
               const float* __restrict__ cb, float* __restrict__ xc) {
  const int idx = blockIdx.x * 256 + threadIdx.x;    // over B*L*CONVDIM
  const int c  = idx % CONVDIM;
  const int bl = idx / CONVDIM;
  const int l  = bl % SEQLEN;
  float acc = cb[c];
#pragma unroll
  for (int t = 0; t < 4; ++t) {
    const int ls = l - 3 + t;
    if (ls >= 0)
      acc += cw[c * 4 + t] * zx[(size_t)(bl - l + ls) * DINPROJ + DINNER + c];
  }
  xc[(size_t)bl * CONVDIM + c] = acc / (1.0f + expf(-acc));
}

// ---------------------------------------------------------------------------
// Selective scan: one block per (head, batch). 256 threads; thread t owns
// state rows p=t>>2, n-range (t&3)*32..+31  (32 fp32 state regs / thread).
// ---------------------------------------------------------------------------
__global__ __launch_bounds__(256)
void ssm_scan(const float* __restrict__ zx, const float* __restrict__ xc,
              float* __restrict__ y, const float* __restrict__ dt_bias,
              const float* __restrict__ A_log, const float* __restrict__ Dp) {
  __shared__ float sB[DSTATE];
  __shared__ float sC[DSTATE];
  const int h   = blockIdx.x;
  const int b   = blockIdx.y;
  const int tid = threadIdx.x;
  const int p   = tid >> 2;
  const int nq  = tid & 3;
  const float An  = -expf(A_log[h]);
  const float Dh  = Dp[h];
  const float dtb = dt_bias[h];

  float hs[32];
#pragma unroll
  for (int j = 0; j < 32; ++j) hs[j] = 0.0f;

  const float4* sB4 = (const float4*)sB;
  const float4* sC4 = (const float4*)sC;

  for (int l = 0; l < SEQLEN; ++l) {
    const size_t rowc = (size_t)(b * SEQLEN + l) * CONVDIM;
    if (tid < 128) sB[tid]       = xc[rowc + DINNER + tid];
    else           sC[tid - 128] = xc[rowc + DINNER + DSTATE + (tid - 128)];
    const float x_p   = xc[rowc + h * HEADDIM + p];
    const float dtraw = zx[(size_t)(b * SEQLEN + l) * DINPROJ +
                           (2 * DINNER + 2 * DSTATE) + h] + dtb;
    const float dtv = (dtraw > 20.0f) ? dtraw : log1pf(expf(dtraw));
    const float dA  = expf(dtv * An);
    const float dtx = dtv * x_p;
    __syncthreads();

    float partial = 0.0f;
#pragma unroll
    for (int j = 0; j < 8; ++j) {
      const float4 Bv = sB4[nq * 8 + j];
      const float4 Cv = sC4[nq * 8 + j];
      float* hp = &hs[j * 4];
      hp[0] = hp[0] * dA + dtx * Bv.x; partial += hp[0] * Cv.x;
      hp[1] = hp[1] * dA + dtx * Bv.y; partial += hp[1] * Cv.y;
      hp[2] = hp[2] * dA + dtx * Bv.z; partial += hp[2] * Cv.z;
      hp[3] = hp[3] * dA + dtx * Bv.w; partial += hp[3] * Cv.w;
    }
    // reduce the 4 n-slices (consecutive lanes in one wave32)
    partial += __shfl_xor(partial, 1, 32);
    partial += __shfl_xor(partial, 2, 32);
    if (nq == 0)
      y[(size_t)(b * SEQLEN + l) * DINNER + h * HEADDIM + p] = partial + Dh * x_p;
    __syncthreads();
  }
}

// ---------------------------------------------------------------------------
// y = y * silu(z); RMS-normalize over DINNER; scale by norm_w. In-place on y.
// One block (256 threads) per (b,l) row; 8 elements per thread.
// ---------------------------------------------------------------------------
__global__ __launch_bounds__(256)
void gated_rmsnorm(const float* __restrict__ zx, float* __restrict__ y,
                   const float* __restrict__ nw) {
  const int row = blockIdx.x;
  const int tid = threadIdx.x;
  float yv[8];
  float acc = 0.0f;
#pragma unroll
  for (int e = 0; e < 8; ++e) {
    const int i = tid + e * 256;
    const float z = zx[(size_t)row * DINPROJ + i];
    const float s = z / (1.0f + expf(-z));
    const float v = y[(size_t)row * DINNER + i] * s;
    yv[e] = v;
    acc += v * v;
  }
#pragma unroll
  for (int off = 16; off > 0; off >>= 1) acc += __shfl_xor(acc, off, 32);
  __shared__ float red[8];
  if ((tid & 31) == 0) red[tid >> 5] = acc;
  __syncthreads();
  float tot = 0.0f;
#pragma unroll
  for (int w = 0; w < 8; ++w) tot += red[w];
  const float scale = rsqrtf(tot / (float)DINNER + 1e-5f);
#pragma unroll
  for (int e = 0; e < 8; ++e) {
    const int i = tid + e * 256;
    y[(size_t)row * DINNER + i] = yv[e] * scale * nw[i];
  }
}

// ---------------------------------------------------------------------------
// out = 0.5 * (out_f + reverse_L(out_b))
// ---------------------------------------------------------------------------
__global__ __launch_bounds__(256)
void combine(const float* __restrict__ of, const float* __restrict__ ob,
             float* __restrict__ out) {
  const int idx = blockIdx.x * 256 + threadIdx.x;
  const int m  = idx % DMODEL;
  const int bl = idx / DMODEL;
  const int l  = bl % SEQLEN;
  const int b  = bl / SEQLEN;
  const size_t rb = (size_t)(b * SEQLEN + (SEQLEN - 1 - l)) * DMODEL + m;
  out[idx] = 0.5f * (of[idx] + ob[rb]);
}

// ---------------------------------------------------------------------------
extern "C" void kernel_launch(void* const* d_in, const int* in_sizes, int n_in,
                              void* d_out, int out_size, void* d_ws, size_t ws_size,
                              hipStream_t stream) {
  const float* u = (const float*)d_in[0];
  float* ws = (float*)d_ws;

  const size_t SZ_ZX = (size_t)BATCH * SEQLEN * DINPROJ;
  const size_t SZ_XC = (size_t)BATCH * SEQLEN * CONVDIM;
  const size_t SZ_Y  = (size_t)BATCH * SEQLEN * DINNER;
  const size_t SZ_O  = (size_t)BATCH * SEQLEN * DMODEL;

  float* zx  = ws;                 // reused by both directions
  float* xc  = zx + SZ_ZX;
  float* yb  = xc + SZ_XC;
  float* o_f = yb + SZ_Y;
  float* o_b = o_f + SZ_O;

  const int M = BATCH * SEQLEN;

  for (int dir = 0; dir < 2; ++dir) {
    const int base = 1 + dir * 8;
    const float* Wi   = (const float*)d_in[base + 0];
    const float* cw   = (const float*)d_in[base + 1];
    const float* cb   = (const float*)d_in[base + 2];
    const float* dtb  = (const float*)d_in[base + 3];
    const float* Alog = (const float*)d_in[base + 4];
    const float* Dp   = (const float*)d_in[base + 5];
    const float* nw   = (const float*)d_in[base + 6];
    const float* Wo   = (const float*)d_in[base + 7];
    float* odir = dir ? o_b : o_f;

    dim3 g1(M / BM, DINPROJ / BN);          // 64 x 137
    gemm_f32_wmma<<<g1, 256, 0, stream>>>(u, Wi, zx, M, DINPROJ, DMODEL,
                                          dir ? SEQLEN : 0);

    const int nconv = BATCH * SEQLEN * CONVDIM;
    conv_silu<<<nconv / 256, 256, 0, stream>>>(zx, cw, cb, xc);

    ssm_scan<<<dim3(NHEADS, BATCH), 256, 0, stream>>>(zx, xc, yb, dtb, Alog, Dp);

    gated_rmsnorm<<<M, 256, 0, stream>>>(zx, yb, nw);

    dim3 g2(M / BM, DMODEL / BN);           // 64 x 32
    gemm_f32_wmma<<<g2, 256, 0, stream>>>(yb, Wo, odir, M, DMODEL, DINNER, 0);
  }

  combine<<<(int)(SZ_O / 256), 256, 0, stream>>>(o_f, o_b, (float*)d_out);
}